// Model_55585466745170
// MI455X (gfx1250) — compile-verified
//
#include <hip/hip_runtime.h>

typedef __attribute__((ext_vector_type(2))) float v2f;
typedef __attribute__((ext_vector_type(8))) float v8f;
typedef __attribute__((ext_vector_type(4))) unsigned int u32x4;
typedef __attribute__((ext_vector_type(8))) int i32x8;
typedef __attribute__((ext_vector_type(4))) int i32x4;

#define D_SINE 200   // 4 * N_FREQ
#define D_EMB  10
#define N_FREQ 50

// ---------------------------------------------------------------------------
// Kernel 1: init out[c,g] = b2[genes_oi[g]]; zero gene histogram
// ---------------------------------------------------------------------------
__global__ void init_kernel(float* __restrict__ out, const float* __restrict__ b2,
                            const int* __restrict__ genes_oi, int out_size,
                            int gene_n, int* __restrict__ counts, int n_genes) {
    int i = blockIdx.x * blockDim.x + threadIdx.x;
    if (i < out_size) out[i] = b2[genes_oi[i % gene_n]];
    if (i < n_genes)  counts[i] = 0;
}

// ---------------------------------------------------------------------------
// Kernel 2: histogram of fragments per gene
// ---------------------------------------------------------------------------
__global__ void hist_kernel(const int* __restrict__ genemapping,
                            int* __restrict__ counts, int n_frag) {
    int i = blockIdx.x * blockDim.x + threadIdx.x;
    if (i < n_frag) atomicAdd(&counts[genemapping[i]], 1);
}

// ---------------------------------------------------------------------------
// Kernel 3: serial exclusive scan over n_genes (tiny: ~500 adds)
// ---------------------------------------------------------------------------
__global__ void scan_kernel(const int* __restrict__ counts, int* __restrict__ offs,
                            int* __restrict__ cursor, int n_genes) {
    if (threadIdx.x == 0 && blockIdx.x == 0) {
        int s = 0;
        for (int g = 0; g < n_genes; ++g) {
            offs[g] = s; cursor[g] = s; s += counts[g];
        }
        offs[n_genes] = s;
    }
}

// ---------------------------------------------------------------------------
// Kernel 4: scatter fragment ids into gene-sorted order
// ---------------------------------------------------------------------------
__global__ void scatter_kernel(const int* __restrict__ genemapping,
                               int* __restrict__ cursor, int* __restrict__ sorted,
                               int n_frag) {
    int i = blockIdx.x * blockDim.x + threadIdx.x;
    if (i < n_frag) {
        int p = atomicAdd(&cursor[genemapping[i]], 1);
        sorted[p] = i;
    }
}

// ---------------------------------------------------------------------------
// Kernel 5: per-gene GEMM via V_WMMA_F32_16X16X4_F32.
// One workgroup per gene; gene weights [200x10] f32 DMA'd into LDS by the
// Tensor Data Mover (TDM) as a 2D tile (D# built inline, wave 0 issues,
// s_wait_tensorcnt + barrier). K=200 swept in 50 WMMA steps of K=4.
// A-matrix f32 16x4 layout: lane<16 holds (K+0,K+1) for M=lane,
//                           lane>=16 holds (K+2,K+3) for M=lane-16.
// C layout: acc[r] = (M = r or 8+r, N = lane&15). Columns n>=10 are garbage
// by construction (B column clamped) and are never read.
// ---------------------------------------------------------------------------
__global__ __launch_bounds__(128) void frag_wmma_kernel(
    const float* __restrict__ coords, const float* __restrict__ w1,
    const float* __restrict__ b1, const float* __restrict__ w2,
    const int* __restrict__ cellxgene, const int* __restrict__ genes_oi,
    const int* __restrict__ offs, const int* __restrict__ sorted,
    float* __restrict__ out, int gene_n) {

    __shared__ float Bsh[D_SINE * D_EMB];   // compact [s][d] tile, 8000 B
    __shared__ float Fsh[N_FREQ];           // frequency table

    const int g = blockIdx.x;

    // --- TDM: DMA w1[g] (200 rows x 10 f32, row stride 10) into LDS ---
    if (threadIdx.x < 32) {
        unsigned long long gaddr =
            (unsigned long long)(uintptr_t)(w1 + (size_t)g * D_SINE * D_EMB);
        unsigned int ldsoff = (unsigned int)(uintptr_t)(void*)&Bsh[0];

        u32x4 g0;
        g0.x = 1u;                                        // count=1, flags=0
        g0.y = ldsoff;                                    // lds_addr [63:32]
        g0.z = (unsigned int)(gaddr & 0xffffffffull);     // global_addr [95:64]
        g0.w = (unsigned int)((gaddr >> 32) & 0x01ffffffull)
             | (2u << 30);                                // addr[56:32] | type=2

        i32x8 g1;
        g1[0] = (int)(2u << 16);          // data_size=2 (4 bytes); mask/flags=0
        g1[1] = (int)((unsigned)D_EMB << 16);   // tensor_dim0 low16 @ [63:48]
        g1[2] = (int)((unsigned)D_SINE << 16);  // tensor_dim1 low16 @ [95:80]
        g1[3] = (int)((unsigned)D_EMB << 16);   // tile_dim0 @ [127:112]
        g1[4] = D_SINE;                          // tile_dim1 @ [143:128]
        g1[5] = D_EMB;                           // tensor_dim0_stride low32
        g1[6] = 0;
        g1[7] = 0;

        i32x4 gz4 = {0, 0, 0, 0};                // groups 2/3 unused (2D)
        i32x8 gz8 = {0, 0, 0, 0, 0, 0, 0, 0};    // trailing group (unused)
        __builtin_amdgcn_tensor_load_to_lds(g0, g1, gz4, gz4, gz8, 0);
        __builtin_amdgcn_s_wait_tensorcnt(0);
    }
    for (int i = threadIdx.x; i < N_FREQ; i += blockDim.x) {
        // base[i] = 1000^(-2*(i+1)/N_FREQ) = exp(-(i+1)*(2/N_FREQ)*ln(1000))
        Fsh[i] = __expf(-(float)(i + 1) * (2.0f / (float)N_FREQ) * 6.907755279f);
    }
    __syncthreads();

    const int lane  = threadIdx.x & 31;
    const int wave  = threadIdx.x >> 5;
    const int nwave = blockDim.x >> 5;
    const int n     = lane & 15;                        // N / column index
    const int nn    = (n < D_EMB) ? n : (D_EMB - 1);    // clamp: cols >=10 unused
    const int khalf = (lane < 16) ? 0 : 2;

    const int start  = offs[g];
    const int end    = offs[g + 1];
    const int ntiles = (end - start + 15) >> 4;

    const float biasn = (n < D_EMB) ? b1[g * D_EMB + n] : 0.0f;

    for (int t = wave; t < ntiles; t += nwave) {
        const int  base   = start + t * 16;
        const bool avalid = (base + n) < end;           // fragment M = lane&15
        const int  fid    = avalid ? sorted[base + n] : 0;
        const float cx    = coords[2 * fid];            // garbage rows unused
        const float cy    = coords[2 * fid + 1];
        const int  ixv    = avalid ? cellxgene[fid] : -1;

        v8f acc = {};
        for (int kb = 0; kb < D_SINE; kb += 4) {
            const int   k0  = kb + khalf;               // even; k0,k0+1 share freq
            const float c   = (k0 < 2 * N_FREQ) ? cx : cy;
            const int   j   = (k0 < 2 * N_FREQ) ? k0 : (k0 - 2 * N_FREQ);
            const float ang = c * Fsh[j >> 1];
            v2f a, b;
            a.x = __sinf(ang);                          // phase 0
            a.y = __cosf(ang);                          // phase pi/2
            b.x = Bsh[k0 * D_EMB + nn];
            b.y = Bsh[(k0 + 1) * D_EMB + nn];
            acc = __builtin_amdgcn_wmma_f32_16x16x4_f32(
                false, a, false, b, (short)0, acc, false, false);
        }

        // Epilogue: bias + sigmoid + dot(w2[genes_oi[ix % gene_n]]) + atomic
        #pragma unroll
        for (int r = 0; r < 8; ++r) {
            const int ix_lo = __shfl(ixv, r, 32);       // fragment M=r
            const int ix_hi = __shfl(ixv, 8 + r, 32);   // fragment M=8+r
            const int myix  = (lane < 16) ? ix_lo : ix_hi;
            float val = 0.0f;
            if (myix >= 0 && n < D_EMB) {
                const float h  = acc[r] + biasn;
                const float sg = 1.0f / (1.0f + __expf(-h));
                const int   gp = genes_oi[myix % gene_n];
                val = sg * w2[gp * D_EMB + n];
            }
            // reduce across the 16 columns of each half-wave
            val += __shfl_xor(val, 1, 32);
            val += __shfl_xor(val, 2, 32);
            val += __shfl_xor(val, 4, 32);
            val += __shfl_xor(val, 8, 32);
            if (lane == 0  && ix_lo >= 0) atomicAdd(&out[ix_lo], val);
            if (lane == 16 && ix_hi >= 0) atomicAdd(&out[ix_hi], val);
        }
    }
}

// ---------------------------------------------------------------------------
extern "C" void kernel_launch(void* const* d_in, const int* in_sizes, int n_in,
                              void* d_out, int out_size, void* d_ws, size_t ws_size,
                              hipStream_t stream) {
    const float* coords      = (const float*)d_in[0];
    const float* w1          = (const float*)d_in[1];
    const float* b1          = (const float*)d_in[2];
    const float* w2          = (const float*)d_in[3];
    const float* b2          = (const float*)d_in[4];
    const int*   genemapping = (const int*)d_in[5];
    const int*   cellxgene   = (const int*)d_in[6];
    const int*   genes_oi    = (const int*)d_in[7];

    const int n_frag  = in_sizes[0] / 2;
    const int n_genes = in_sizes[4];
    const int gene_n  = in_sizes[7];

    int* counts = (int*)d_ws;
    int* offs   = counts + n_genes;
    int* cursor = offs + n_genes + 1;
    int* sorted = cursor + n_genes;

    const int initN = (out_size > n_genes) ? out_size : n_genes;
    init_kernel<<<(initN + 255) / 256, 256, 0, stream>>>(
        (float*)d_out, b2, genes_oi, out_size, gene_n, counts, n_genes);

    hist_kernel<<<(n_frag + 255) / 256, 256, 0, stream>>>(genemapping, counts, n_frag);

    scan_kernel<<<1, 32, 0, stream>>>(counts, offs, cursor, n_genes);

    scatter_kernel<<<(n_frag + 255) / 256, 256, 0, stream>>>(genemapping, cursor,
                                                             sorted, n_frag);

    frag_wmma_kernel<<<n_genes, 128, 0, stream>>>(
        coords, w1, b1, w2, cellxgene, genes_oi, offs, sorted,
        (float*)d_out, gene_n);
}